// PatchShift_47287589929241
// MI455X (gfx1250) — compile-verified
//
#include <hip/hip_runtime.h>
#include <stdint.h>

// PatchShift: x (4096, 4, 49, 32) f32, B = batch(8) x T(8) x G(64).
// out[(B,h,n),c] = in[(B + (t_src - t)*64, h, n), c],
//   t = (B>>6)&7, t_src = (t - shift(n)) & 7,
//   shift(n) = TBL[(n/7)%3][(n%7)%3], TBL = {{-4,1,2},{-1,0,3},{-2,-3,4}}.
// Pure permutation copy: ~205 MB traffic, ~8.8 us roofline @ 23.3 TB/s.
//
// Thread k-stride = 802816 chunks = 512 B-units = 8*64 -> t, n, h invariant
// across k, so the gather index math is computed ONCE per thread and every
// address is base + k*const. Data moves through the gfx1250 async DMA path
// (global -> LDS -> global, ASYNCcnt-tracked) in B128 ops, GVS addressing.

#define BLOCK 256
#define CPT   8   // 16B chunks per thread

static constexpr int TOTAL_CHUNKS = 6422528;  // 4096*4*49*32 / 4
static constexpr int NBLOCKS      = 3136;     // TOTAL_CHUNKS / (BLOCK*CPT), exact
static constexpr int CHUNKS_PER_B = 1568;     // 4*49*8 chunks per unit step of B
static constexpr int KSTRIDE      = NBLOCKS * BLOCK;          // 802816 chunks
static constexpr unsigned KSTRIDE_BYTES = (unsigned)KSTRIDE * 16u;  // 12,845,056

__device__ __forceinline__ int src_chunk_of(int chunk) {
  int row = chunk >> 3;          // (B*4 + h)*49 + n
  int n   = row % 49;
  int bh  = row / 49;            // B*4 + h
  int B   = bh >> 2;
  int t   = (B >> 6) & 7;
  int i   = n / 7;               // 0..6
  int j   = n - i * 7;           // 0..6
  int idx = (i % 3) * 3 + (j % 3);
  // nibble-packed (shift+4) for TBL rows {-4,1,2},{-1,0,3},{-2,-3,4}
  const unsigned long long P = 0x812743650ull;
  int s    = (int)((P >> (idx * 4)) & 0xFull) - 4;
  int tsrc = (t - s) & 7;        // mod 8, safe for negatives
  return chunk + (tsrc - t) * 64 * CHUNKS_PER_B;
}

__global__ __launch_bounds__(BLOCK)
void PatchShift_kernel(const float* __restrict__ in, float* __restrict__ out) {
  const int tid0 = blockIdx.x * BLOCK + threadIdx.x;  // k=0 chunk, < 802816

#if defined(__HIP_DEVICE_COMPILE__)
  __shared__ __align__(16) char smem[BLOCK * 16 * CPT];  // 32 KB/block

  // One-time gather index math (t,n,h invariant across k).
  const unsigned src0 = (unsigned)src_chunk_of(tid0) * 16u;  // in [0, 12.8MB)
  const unsigned dst0 = (unsigned)tid0 * 16u;

  // Wave-relative LDS byte address of this thread's slot 0.
  const unsigned lds0 =
      (unsigned)(uintptr_t)(__attribute__((address_space(3))) char*)smem +
      (unsigned)threadIdx.x * 16u;

#pragma unroll
  for (int k = 0; k < CPT; ++k) {
    unsigned la = lds0 + (unsigned)k * (BLOCK * 16u);
    unsigned va = src0 + (unsigned)k * KSTRIDE_BYTES;
    // GVS: mem = SGPR64(in) + u32 voffset. VDST = LDS addr. ASYNCcnt-tracked.
    asm volatile("global_load_async_to_lds_b128 %0, %1, %2"
                 :: "v"(la), "v"(va), "s"(in) : "memory");
  }

  // All this wave's async loads complete -> data resident in LDS.
  asm volatile("s_wait_asynccnt 0x0" ::: "memory");

#pragma unroll
  for (int k = 0; k < CPT; ++k) {
    unsigned la = lds0 + (unsigned)k * (BLOCK * 16u);
    unsigned va = dst0 + (unsigned)k * KSTRIDE_BYTES;
    // GVS: mem = SGPR64(out) + u32 voffset. VSRC = LDS addr.
    asm volatile("global_store_async_from_lds_b128 %0, %1, %2"
                 :: "v"(va), "v"(la), "s"(out) : "memory");
  }
  // S_ENDPGM performs an implicit wait-idle -> async stores drain before
  // the workgroup's LDS is deallocated.
#else
  (void)tid0; (void)in; (void)out;
#endif
}

extern "C" void kernel_launch(void* const* d_in, const int* in_sizes, int n_in,
                              void* d_out, int out_size, void* d_ws, size_t ws_size,
                              hipStream_t stream) {
  const float* x = (const float*)d_in[0];
  float* out     = (float*)d_out;
  // 3136 blocks * 256 threads * 8 chunks * 16B == 102.76 MB == whole tensor, exact.
  PatchShift_kernel<<<dim3(NBLOCKS), dim3(BLOCK), 0, stream>>>(x, out);
}